// Conv2D_6674379178255
// MI455X (gfx1250) — compile-verified
//
#include <hip/hip_runtime.h>
#include <hip/hip_bf16.h>
#include <stdint.h>

// ---- vector types ----
typedef __attribute__((ext_vector_type(2))) float    v2f;
typedef __attribute__((ext_vector_type(8))) float    v8f;
typedef __attribute__((ext_vector_type(4))) uint32_t u32x4;
typedef __attribute__((ext_vector_type(8))) int32_t  i32x8;
typedef __attribute__((ext_vector_type(4))) int32_t  i32x4;

#define HH 4096
#define WW 4096
#define KH 7
#define KW 7
#define OH 4090
#define OW 4090

#define BLOCK_COLS 512                    // output cols per block
#define BLOCK_ROWS 16                     // output rows per block
#define LDS_W (BLOCK_COLS + KW - 1)       // 518 floats per LDS row
#define LDS_H (BLOCK_ROWS + KH - 1)       // 22 LDS rows
#define NWAVES 8
#define WAVE_COLS (BLOCK_COLS / NWAVES)   // 64
#define TILES_PER_WAVE (WAVE_COLS / 16)   // 4
#define KCHUNKS 6                         // ceil(22 / 4) K-chunks of the banded GEMM

__global__ __launch_bounds__(256) void conv7x7_wmma(
    const float* __restrict__ x, const float* __restrict__ wgt,
    const float* __restrict__ bias, float* __restrict__ out)
{
    __shared__ float xs[LDS_H * LDS_W];   // 45,584 B input patch
    __shared__ float wls[KH * KW];

    const int tid  = threadIdx.x;
    const int lane = tid & 31;
    const int wave = tid >> 5;
    const int n    = lane & 15;           // N index (output col within tile)
    const int hsel = lane >> 4;           // lane half: selects K pair / M+8

    const int c0 = blockIdx.x * BLOCK_COLS;   // input & output col origin
    const int r0 = blockIdx.y * BLOCK_ROWS;   // input & output row origin

    // stage the 49 weights into LDS (broadcast source for B fragments)
    if (tid < KH * KW) wls[tid] = wgt[tid];

    // ---- TDM: one tensor_load_to_lds per block (wave 0 issues it) ----
    if (wave == 0) {
        // flat LDS address truncates to the raw LDS byte offset (aperture in high bits)
        uint32_t lds_base = (uint32_t)(uintptr_t)(&xs[0]);
        uint64_t gaddr = (uint64_t)(uintptr_t)x + (((uint64_t)r0 * WW + (uint64_t)c0) << 2);
        // tensor dims are relative to the tile start -> OOB rows/cols zero-fill
        uint32_t tdim0 = (uint32_t)(WW - c0);
        uint32_t tdim1 = (uint32_t)(HH - r0);

        // D# group 0: count=1 | lds_addr | global_addr[56:0] | type=2
        u32x4 g0 = { 1u,
                     lds_base,
                     (uint32_t)gaddr,
                     (uint32_t)((gaddr >> 32) & 0x01FFFFFFull) | (2u << 30) };
        // D# group 1: data_size=4B; tensor_dim0/1; tile 518 x 22; stride0 = 4096
        i32x8 g1 = { (int32_t)(2u << 16),                                  // data_size=2 (4B)
                     (int32_t)((tdim0 & 0xFFFFu) << 16),                   // tdim0[15:0]
                     (int32_t)((tdim0 >> 16) | ((tdim1 & 0xFFFFu) << 16)), // tdim0[31:16]|tdim1[15:0]
                     (int32_t)((tdim1 >> 16) | ((uint32_t)LDS_W << 16)),   // tdim1[31:16]|tile_dim0
                     (int32_t)LDS_H,                                       // tile_dim1=22, tile_dim2=0
                     (int32_t)WW,                                          // tensor_dim0_stride
                     0, 0 };
        i32x4 gz4 = { 0, 0, 0, 0 };
        i32x8 gz8 = { 0, 0, 0, 0, 0, 0, 0, 0 };
        __builtin_amdgcn_tensor_load_to_lds(g0, g1, gz4, gz4, gz8, 0);
        __builtin_amdgcn_s_wait_tensorcnt(0);
    }
    __syncthreads();

    // ---- precompute banded-Toeplitz weight (B) fragments, once per wave ----
    // B[j][n] = w[kh][j-n] for 0 <= j-n < 7, else 0.  K layout guess mirrors A:
    // VGPR v holds K = v + 2*hsel within each K=4 chunk.
    float bfrag[KH][KCHUNKS][2];
    #pragma unroll
    for (int kh = 0; kh < KH; ++kh)
        #pragma unroll
        for (int ch = 0; ch < KCHUNKS; ++ch)
            #pragma unroll
            for (int v = 0; v < 2; ++v) {
                int d = 4 * ch + v + 2 * hsel - n;     // j - n
                bfrag[kh][ch][v] = (d >= 0 && d < KW) ? wls[kh * KW + d] : 0.0f;
            }

    const float bv = bias[0];

    // ---- 4 output tiles of 16x16 per wave ----
    for (int t = 0; t < TILES_PER_WAVE; ++t) {
        const int cbase = wave * WAVE_COLS + t * 16;   // LDS col of tile origin

        v8f acc;
        #pragma unroll
        for (int v = 0; v < 8; ++v) acc[v] = bv;

        #pragma unroll
        for (int kh = 0; kh < KH; ++kh) {
            // A[m][k]: lane m = n, row m+kh, cols cbase + 4*ch + (2*hsel + {0,1})
            const float* rowp = &xs[(n + kh) * LDS_W + cbase + 2 * hsel];
            #pragma unroll
            for (int ch = 0; ch < KCHUNKS; ++ch) {
                v2f a = *(const v2f*)(rowp + 4 * ch);  // 8B-aligned -> ds_load_b64
                v2f b; b[0] = bfrag[kh][ch][0]; b[1] = bfrag[kh][ch][1];
                acc = __builtin_amdgcn_wmma_f32_16x16x4_f32(
                    false, a, false, b, (short)0, acc, false, false);
            }
        }

        // C/D layout: VGPR v -> row r0+v (lanes 0-15) / r0+v+8 (lanes 16-31), col = n
        #pragma unroll
        for (int v = 0; v < 8; ++v) {
            int row = r0 + v + 8 * hsel;
            int col = c0 + cbase + n;
            if (row < OH && col < OW) out[row * OW + col] = acc[v];
        }
    }
}

extern "C" void kernel_launch(void* const* d_in, const int* in_sizes, int n_in,
                              void* d_out, int out_size, void* d_ws, size_t ws_size,
                              hipStream_t stream) {
    const float* x    = (const float*)d_in[0];
    const float* wgt  = (const float*)d_in[1];
    const float* bias = (const float*)d_in[2];
    float* out = (float*)d_out;
    (void)in_sizes; (void)n_in; (void)out_size; (void)d_ws; (void)ws_size;

    dim3 grid((OW + BLOCK_COLS - 1) / BLOCK_COLS,   // 8
              (OH + BLOCK_ROWS - 1) / BLOCK_ROWS);  // 256
    conv7x7_wmma<<<grid, 256, 0, stream>>>(x, wgt, bias, out);
}